// LinearAttentionFusedRecurrentKernel_18726057411019
// MI455X (gfx1250) — compile-verified
//
#include <hip/hip_runtime.h>

typedef __attribute__((ext_vector_type(2))) float v2f;
typedef __attribute__((ext_vector_type(8))) float v8f;

#define B_ 4
#define S_ 2048
#define H_ 16
#define D_ 128
#define C_ 16
#define NCHUNK (S_ / C_)
#define TSTRIDE (H_ * D_)        // elements between consecutive timesteps
#define CSTRIDE (C_ * TSTRIDE)   // elements between consecutive chunks

__device__ __forceinline__ size_t gidx(int b, int t, int h, int d) {
    return ((size_t)((b * S_ + t) * H_ + h)) * (size_t)D_ + (size_t)d;
}

__device__ __forceinline__ v8f wmma4(v2f a, v2f b, v8f c) {
    // D = A(16x4, f32) * B(4x16, f32) + C(16x16, f32)
    return __builtin_amdgcn_wmma_f32_16x16x4_f32(
        /*neg_a=*/false, a, /*neg_b=*/false, b,
        /*c_mod=*/(short)0, c, /*reuse_a=*/false, /*reuse_b=*/false);
}

// One wave32 per (b, h, 16-wide Dv column group).
// Chunked linear attention, chunk = 16 timesteps:
//   S' = causal_mask(Q K^T); o = scale*(S' V + Q H_prev); H += K^T V
__launch_bounds__(32)
__global__ void lin_attn_chunked_kernel(const float* __restrict__ q,
                                        const float* __restrict__ k,
                                        const float* __restrict__ v,
                                        float* __restrict__ out) {
    __shared__ float sS[C_][C_];   // masked Q K^T tile   (1 KB)
    __shared__ float sK[C_][D_];   // K chunk             (8 KB)
    __shared__ float sH[D_][16];   // state slice H[:,16] (8 KB)

    const int lane = (int)(threadIdx.x & 31u);
    const int lo = lane & 15;   // row / column index within 16
    const int hi = lane >> 4;   // which half of the wave

    int blk = (int)blockIdx.x;
    const int jg = blk & 7;             // Dv column group
    blk >>= 3;
    const int h = blk & (H_ - 1);
    const int b = blk >> 4;
    const int j0 = jg * 16;

    const float scale = 0.08838834764831845f; // 128^-0.5
    const v8f vzero = (v8f){0.f, 0.f, 0.f, 0.f, 0.f, 0.f, 0.f, 0.f};

    // Per-lane base pointers; advanced by one chunk stride each iteration.
    const float* qrow = q + gidx(b, lo, h, 2 * hi);          // A-frag row of Q
    const float* krow = k + gidx(b, lo, h, 2 * hi);          // B-frag "row" of K^T
    const float* vcol = v + gidx(b, 0, h, j0 + lo);          // B-frag column of V
    float*       orow = out + gidx(b, 0, h, j0 + lo);        // C-layout output column

    // State slice H[0:128, j0:j0+16] as 8 WMMA C/D tiles (C layout):
    // tile r: element (m = 16r + p + 8*hi, n = lo) in vgpr p.
    v8f htile[8];
#pragma unroll
    for (int r = 0; r < 8; ++r) htile[r] = vzero;

    // Persistent (unscaled) Q fragments for the current chunk.
    v2f qf[32];

    for (int c = 0; c < NCHUNK; ++c) {
        __syncthreads(); // protect sK/sS/sH reuse across chunks

        // ---- Load Q and K; compute S' = Q K^T (2 accumulators); stage K ----
        // A layout (16x4 f32): lane holds row M = lo, K = 4f + 2*hi + {0,1}
        // B layout for K^T  : lane holds col N = lo, K = 4f + 2*hi + {0,1}
        v8f s0 = vzero, s1 = vzero;
#pragma unroll
        for (int fb = 0; fb < 4; ++fb) {
            v2f kv[8];
#pragma unroll
            for (int u = 0; u < 8; ++u) {
                const int f = fb * 8 + u;
                qf[f]  = *(const v2f*)(qrow + 4 * f);
                kv[u]  = *(const v2f*)(krow + 4 * f);
            }
#pragma unroll
            for (int u = 0; u < 8; ++u) {
                const int f = fb * 8 + u;
                *(v2f*)(&sK[lo][4 * f + 2 * hi]) = kv[u];
            }
#pragma unroll
            for (int u = 0; u < 8; ++u) {
                const int f = fb * 8 + u;
                if (u & 1) s1 = wmma4(qf[f], kv[u], s1);
                else       s0 = wmma4(qf[f], kv[u], s0);
            }
        }
        v8f s = s0 + s1;

        // Prefetch next chunk (global_prefetch_b8)
        if (c + 1 < NCHUNK) {
            __builtin_prefetch(qrow + CSTRIDE, 0, 1);
            __builtin_prefetch(krow + CSTRIDE, 0, 1);
            __builtin_prefetch(vcol + CSTRIDE, 0, 1);
        }

        // ---- Causal mask (diagonal kept) ----
        // C layout: element (m = p + 8*hi, n = lo) in vgpr p; keep n <= m.
#pragma unroll
        for (int p = 0; p < 8; ++p) {
            const int m = p + 8 * hi;
            s[p] = (lo > m) ? 0.0f : s[p];
        }

        // ---- Stage S' and H_prev to LDS for layout conversion ----
#pragma unroll
        for (int p = 0; p < 8; ++p) sS[p + 8 * hi][lo] = s[p];
#pragma unroll
        for (int r = 0; r < 8; ++r) {
#pragma unroll
            for (int p = 0; p < 8; ++p) sH[16 * r + p + 8 * hi][lo] = htile[r][p];
        }
        __syncthreads();

        // ---- Intra-chunk: o += S'_masked @ V ----
        v2f vfrag[4], sa[4];
#pragma unroll
        for (int g = 0; g < 4; ++g) {
            const int t = 4 * g + 2 * hi;
            vfrag[g].x = vcol[t * TSTRIDE];
            vfrag[g].y = vcol[(t + 1) * TSTRIDE];
            sa[g] = *(const v2f*)(&sS[lo][t]);
        }
        v8f o0 = vzero, o1 = vzero;
#pragma unroll
        for (int g = 0; g < 4; ++g) {
            if (g & 1) o1 = wmma4(sa[g], vfrag[g], o1);
            else       o0 = wmma4(sa[g], vfrag[g], o0);
        }

        // ---- Inter-chunk: o += Q @ H_prev (batched LDS frag loads) ----
#pragma unroll
        for (int fb = 0; fb < 4; ++fb) {
            v2f hb[8];
#pragma unroll
            for (int u = 0; u < 8; ++u) {
                const int f = fb * 8 + u;
                hb[u].x = sH[4 * f + 2 * hi][lo];
                hb[u].y = sH[4 * f + 2 * hi + 1][lo];
            }
#pragma unroll
            for (int u = 0; u < 8; ++u) {
                const int f = fb * 8 + u;
                if (u & 1) o1 = wmma4(qf[f], hb[u], o1);
                else       o0 = wmma4(qf[f], hb[u], o0);
            }
        }
        v8f o = (o0 + o1) * scale;

        // ---- Store o (C layout: row t0 + p + 8*hi, col j0 + lo) ----
#pragma unroll
        for (int p = 0; p < 8; ++p) {
            orow[(p + 8 * hi) * TSTRIDE] = o[p];
        }

        // ---- State update: H += K^T V (8 independent chains, batched loads) ----
#pragma unroll
        for (int rb = 0; rb < 4; ++rb) {
            v2f ka[8];
#pragma unroll
            for (int r2 = 0; r2 < 2; ++r2) {
                const int r = 2 * rb + r2;
#pragma unroll
                for (int g = 0; g < 4; ++g) {
                    const int t = 4 * g + 2 * hi;
                    ka[r2 * 4 + g].x = sK[t][16 * r + lo];
                    ka[r2 * 4 + g].y = sK[t + 1][16 * r + lo];
                }
            }
#pragma unroll
            for (int r2 = 0; r2 < 2; ++r2) {
                const int r = 2 * rb + r2;
#pragma unroll
                for (int g = 0; g < 4; ++g) {
                    htile[r] = wmma4(ka[r2 * 4 + g], vfrag[g], htile[r]);
                }
            }
        }

        // ---- Advance chunk pointers ----
        qrow += CSTRIDE;
        krow += CSTRIDE;
        vcol += CSTRIDE;
        orow += CSTRIDE;
    }
}

extern "C" void kernel_launch(void* const* d_in, const int* in_sizes, int n_in,
                              void* d_out, int out_size, void* d_ws, size_t ws_size,
                              hipStream_t stream) {
    (void)in_sizes; (void)n_in; (void)out_size; (void)d_ws; (void)ws_size;
    const float* q = (const float*)d_in[0];
    const float* k = (const float*)d_in[1];
    const float* v = (const float*)d_in[2];
    float* out = (float*)d_out;

    dim3 grid(B_ * H_ * (D_ / 16)); // 512 single-wave workgroups
    dim3 block(32);
    lin_attn_chunked_kernel<<<grid, block, 0, stream>>>(q, k, v, out);
}